// CTCLayer_39728447488221
// MI455X (gfx1250) — compile-verified
//
#include <hip/hip_runtime.h>

#define BB 128
#define TT 512
#define CC 1024
#define LL 48
#define SS 97      // 2*LL+1 extended states
#define SPAD 128   // padded state stride (32 lanes * 4 states)
#define EPSF 1e-7f
#define NEGV (-1e30f)
#define DEPTH 24   // async-to-LDS ring depth (s_wait_asynccnt DEPTH-1)

#define LOG2E 1.4426950408889634f
#define LN2   0.6931471805599453f

// logaddexp on the dependent chain: raw v_exp_f32 + v_log_f32 (both base-2).
// d = min-max <= 0, so 2^(d*log2e) is in [0,1]; exact 0 for the -1e30 surrogate.
__device__ __forceinline__ float lae(float a, float b) {
    float mx = fmaxf(a, b);
    float d  = fminf(a, b) - mx;
    float e  = __builtin_amdgcn_exp2f(d * LOG2E);          // v_exp_f32
    return mx + LN2 * __builtin_amdgcn_logf(1.0f + e);     // v_log_f32
}

// ---------------- Phase 1: gather label columns + log ----------------
// One block per (b,t); thread s produces lp[b,t,s] for the padded S=128 layout.
// Bandwidth-bound: ~210 MB of effective traffic at cacheline granularity.
__global__ __launch_bounds__(128) void ctc_gather(const float* __restrict__ y_pred,
                                                  const long long* __restrict__ y_true,
                                                  float* __restrict__ lp) {
    const int bt = blockIdx.x;        // b*TT + t
    const int b  = bt >> 9;           // TT = 512
    const int s  = threadIdx.x;
    int cls = CC - 1;                 // blank
    if ((s & 1) && s < SS) cls = (int)y_true[b * LL + (s >> 1)];
    const float* row = y_pred + ((size_t)bt << 10);   // *CC
    __builtin_prefetch(row + CC, 0, 0);               // global_prefetch_b8 for next row
    float v = row[cls];
    lp[((size_t)bt << 7) + s] = LN2 * __builtin_amdgcn_logf(v + EPSF);
}

// ---------------- Phase 2: sequential alpha scan, one wave32 per batch ----------------
// Lane L holds states s = 4L..4L+3. No barriers: neighbor exchange via __shfl_up,
// lp rows streamed DEPTH ahead into LDS with global_load_async_to_lds_b128.
__global__ __launch_bounds__(128) void ctc_scan(const float* __restrict__ lp,
                                                const long long* __restrict__ y_true,
                                                float* __restrict__ out) {
    __shared__ __align__(16) float ring[4][DEPTH][SPAD];   // 48 KB
    const int lane = threadIdx.x & 31;
    const int wv   = threadIdx.x >> 5;
    const int b    = blockIdx.x * 4 + wv;
    const float* lpb = lp + (size_t)b * TT * SPAD;
    const long long* lab = y_true + b * LL;

    // skip flags: only odd states can skip; s parity == i parity (4L even)
    const int s1 = 4 * lane + 1, s3 = 4 * lane + 3;
    bool skip1 = false, skip3 = false;
    if (s1 < SS) {
        int l1 = (int)lab[s1 >> 1];
        skip1 = (s1 == 1) || (l1 != (int)lab[(s1 >> 1) - 1]);
    }
    if (s3 < SS) {
        int l3 = (int)lab[s3 >> 1];
        skip3 = (l3 != (int)lab[(s3 >> 1) - 1]);
    }

    // alpha at t=0: states 0 and 1 get lp row 0, rest -inf
    float a0 = (lane == 0) ? lpb[0] : NEGV;
    float a1 = (lane == 0) ? lpb[1] : NEGV;
    float a2 = NEGV, a3 = NEGV;

    const uint64_t gbase = (uint64_t)(uintptr_t)lpb + (uint64_t)(lane * 16);

    // prime the ring with rows t = 1..DEPTH (DEPTH < TT-1 here)
    for (int i = 0; i < DEPTH; ++i) {
        uint32_t la = (uint32_t)(uintptr_t)&ring[wv][i][lane * 4];
        uint64_t ga = gbase + (uint64_t)((i + 1) * (SPAD * 4));
        asm volatile("global_load_async_to_lds_b128 %0, %1, off"
                     :: "v"(la), "v"(ga) : "memory");
    }

    int slot = 0;
    for (int t = 1; t < TT; ++t) {
        // oldest of the DEPTH outstanding async loads (in-order per wave) is row t
        asm volatile("s_wait_asynccnt 23" ::: "memory");
        const float4 r = *(const float4*)&ring[wv][slot][lane * 4];

        // previous lane's old state 4L-1; odd states' s-2 is p3 (s=4L+1) or a1 (s=4L+3)
        float p3 = __shfl_up(a3, 1, 32);
        if (lane == 0) p3 = NEGV;

        float n0 = lae(a0, p3) + r.x;                        // even state: no skip
        float n1 = lae(a1, a0); if (skip1) n1 = lae(n1, p3); n1 += r.y;
        float n2 = lae(a2, a1) + r.z;                        // even state: no skip
        float n3 = lae(a3, a2); if (skip3) n3 = lae(n3, a1); n3 += r.w;
        a0 = n0; a1 = n1; a2 = n2; a3 = n3;

        // refill freed slot with row t+DEPTH (clamped dummies keep count uniform)
        int tn = t + DEPTH; if (tn > TT - 1) tn = TT - 1;
        uint32_t la = (uint32_t)(uintptr_t)&ring[wv][slot][lane * 4];
        uint64_t ga = gbase + (uint64_t)(tn * (SPAD * 4));
        asm volatile("global_load_async_to_lds_b128 %0, %1, off"
                     :: "v"(la), "v"(ga) : "memory");
        if (++slot == DEPTH) slot = 0;
    }

    // loss = -logaddexp(alpha[96], alpha[95]); s=96 -> lane24.i0, s=95 -> lane23.i3
    float v96 = __shfl(a0, 24, 32);
    float v95 = __shfl(a3, 23, 32);
    if (lane == 0) out[b] = -lae(v96, v95);
}

extern "C" void kernel_launch(void* const* d_in, const int* in_sizes, int n_in,
                              void* d_out, int out_size, void* d_ws, size_t ws_size,
                              hipStream_t stream) {
    const float*     y_pred = (const float*)d_in[0];        // [B,T,C] f32
    const long long* y_true = (const long long*)d_in[1];    // [B,L] i64
    float* out = (float*)d_out;                             // [B,1] f32
    float* lp  = (float*)d_ws;                              // B*T*SPAD*4 = 32 MiB scratch

    ctc_gather<<<BB * TT, 128, 0, stream>>>(y_pred, y_true, lp);
    ctc_scan<<<BB / 4, 128, 0, stream>>>(lp, y_true, out);

    (void)in_sizes; (void)n_in; (void)out_size; (void)ws_size;
}